// Decagon_34059090657401
// MI455X (gfx1250) — compile-verified
//
#include <hip/hip_runtime.h>
#include <hip/hip_bf16.h>

// ---------------------------------------------------------------------------
// Problem constants (from reference)
// ---------------------------------------------------------------------------
#define ND     600
#define NPRO   19081
#define NSE    964
#define EMB    128
#define FEAT   2048
#define NE_    1200000
#define NPAIRS 8192
#define NANCH  8192
#define NV     (ND + NPRO)

typedef __attribute__((ext_vector_type(2))) float v2f;
typedef __attribute__((ext_vector_type(8))) float v8f;

// ---------------------------------------------------------------------------
// Core fp32 WMMA helper: accumulate C += Arow(len K) x Brow(len K)^T tilewise.
// Per ISA 16x16x4_f32 layout:
//   A frag (2 VGPR): lanes 0-15 hold row m0+l, K = {k, k+1};
//                    lanes 16-31 hold row m0+l, K = {k+2, k+3}
//   B frag mirrors A with W row n0+l.
//   C (8 VGPR): VGPR r -> row m0+r (lanes 0-15) / m0+8+r (lanes 16-31), col n0+lane.
// ---------------------------------------------------------------------------
__device__ __forceinline__ v8f wmma_loop(const float* __restrict__ Ar,
                                         const float* __restrict__ Br,
                                         int K, int kOff, v8f c) {
#pragma unroll 4
    for (int k = 0; k < K; k += 4) {
        const int kb = k + kOff;
        v2f a = { Ar[kb], Ar[kb + 1] };
        v2f b = { Br[kb], Br[kb + 1] };
        c = __builtin_amdgcn_wmma_f32_16x16x4_f32(false, a, false, b,
                                                  (short)0, c, false, false);
    }
    return c;
}

// ---------------------------------------------------------------------------
// Generic out = relu(A(MxK) @ W(NxK)^T + bias), N = 128, out row-stride 128.
// block = 256 threads = 8 waves, wave w covers N-tile w; grid.x = ceil(M/16).
// ---------------------------------------------------------------------------
__global__ void k_gemm_relu(const float* __restrict__ A, int M, int K,
                            const float* __restrict__ W,
                            const float* __restrict__ bias,
                            float* __restrict__ out) {
    const int lane = threadIdx.x & 31;
    const int wave = threadIdx.x >> 5;           // n-tile 0..7
    const int half = lane >> 4;
    const int l    = lane & 15;
    const int m0   = blockIdx.x * 16;
    const int n    = wave * 16 + l;
    int rowA = m0 + l; if (rowA > M - 1) rowA = M - 1;   // clamp (store guarded)
    const int kOff = 2 * half;

    v8f c = {};
    c = wmma_loop(A + (size_t)rowA * K, W + (size_t)n * K, K, kOff, c);

    const float bn = bias[n];
#pragma unroll
    for (int r = 0; r < 8; ++r) {
        const int m = m0 + r + 8 * half;
        if (m < M) out[(size_t)m * EMB + n] = fmaxf(c[r] + bn, 0.f);
    }
}

// ---------------------------------------------------------------------------
// SAGE (only last layer survives in the reference):
//   xd = relu(aggn @ Wl^T + drugF @ Wr^T + bias), M = 600, K = N = 128.
// ---------------------------------------------------------------------------
__global__ void k_sage(const float* __restrict__ aggn,
                       const float* __restrict__ Wl,
                       const float* __restrict__ drugF,
                       const float* __restrict__ Wr,
                       const float* __restrict__ bias,
                       float* __restrict__ xd) {
    const int lane = threadIdx.x & 31;
    const int wave = threadIdx.x >> 5;
    const int half = lane >> 4;
    const int l    = lane & 15;
    const int m0   = blockIdx.x * 16;
    const int n    = wave * 16 + l;
    int rowA = m0 + l; if (rowA > ND - 1) rowA = ND - 1;
    const int kOff = 2 * half;

    v8f c = {};
    c = wmma_loop(aggn  + (size_t)rowA * EMB, Wl + (size_t)n * EMB, EMB, kOff, c);
    c = wmma_loop(drugF + (size_t)rowA * EMB, Wr + (size_t)n * EMB, EMB, kOff, c);

    const float bn = bias[n];
#pragma unroll
    for (int r = 0; r < 8; ++r) {
        const int m = m0 + r + 8 * half;
        if (m < ND) xd[(size_t)m * EMB + n] = fmaxf(c[r] + bn, 0.f);
    }
}

// ---------------------------------------------------------------------------
// Pair MLP: out1 = relu(concat(xd[tpl0], xd[tpl1]) @ outW1(128x256)^T + b).
// The k<128 / k>=128 split is wave-uniform -> two WMMA K=128 passes with
// per-lane gathered A-row bases.
// ---------------------------------------------------------------------------
__global__ void k_pair(const float* __restrict__ xd,
                       const int* __restrict__ tpl,
                       const float* __restrict__ outW1,
                       const float* __restrict__ outb1,
                       float* __restrict__ out1) {
    const int lane = threadIdx.x & 31;
    const int wave = threadIdx.x >> 5;
    const int half = lane >> 4;
    const int l    = lane & 15;
    const int m0   = blockIdx.x * 16;            // NPAIRS divisible by 16
    const int n    = wave * 16 + l;
    const int r    = m0 + l;
    const int i0   = tpl[2 * r];
    const int i1   = tpl[2 * r + 1];
    const int kOff = 2 * half;

    v8f c = {};
    c = wmma_loop(xd + (size_t)i0 * EMB, outW1 + (size_t)n * (2 * EMB),       EMB, kOff, c);
    c = wmma_loop(xd + (size_t)i1 * EMB, outW1 + (size_t)n * (2 * EMB) + EMB, EMB, kOff, c);

    const float bn = outb1[n];
#pragma unroll
    for (int r8 = 0; r8 < 8; ++r8) {
        const int m = m0 + r8 + 8 * half;
        out1[(size_t)m * EMB + n] = fmaxf(c[r8] + bn, 0.f);
    }
}

// ---------------------------------------------------------------------------
// Edge scan: only edges with dst < ND matter. Each wave loads 32 edges,
// ballots qualifying lanes, and for each one all 32 lanes cooperatively
// gather the 128-float source row and atomicAdd into agg[dst]. deg counted too.
// ---------------------------------------------------------------------------
__global__ void k_edges(const int* __restrict__ src, const int* __restrict__ dst,
                        const float* __restrict__ drugF,
                        const float* __restrict__ protEmb,
                        float* __restrict__ agg, float* __restrict__ deg) {
    const int lane   = threadIdx.x & 31;
    const int waveG  = (blockIdx.x * blockDim.x + threadIdx.x) >> 5;
    const int nWaves = (gridDim.x * blockDim.x) >> 5;

    for (int base = waveG * 32; base < NE_; base += nWaves * 32) {
        const int e = base + lane;
        int s = 0, d = -1;
        if (e < NE_) { s = src[e]; d = dst[e]; }
        unsigned mask = __builtin_amdgcn_ballot_w32(d >= 0 && d < ND);
        while (mask) {
            const int i = __builtin_ctz(mask);
            mask &= mask - 1;
            const int ss = __builtin_amdgcn_readlane(s, i);
            const int dd = __builtin_amdgcn_readlane(d, i);
            const float* __restrict__ xrow =
                (ss < ND) ? (drugF + (size_t)ss * EMB)
                          : (protEmb + (size_t)(ss - ND) * EMB);
            float* __restrict__ arow = agg + (size_t)dd * EMB;
#pragma unroll
            for (int q = 0; q < 4; ++q)
                atomicAdd(&arow[lane + 32 * q], xrow[lane + 32 * q]);
            if (lane == 0) atomicAdd(&deg[dd], 1.0f);
        }
    }
}

// agg[v][:] *= 1 / max(deg[v], 1)
__global__ void k_scale(float* __restrict__ agg, const float* __restrict__ deg) {
    const int i = blockIdx.x * blockDim.x + threadIdx.x;
    if (i < ND * EMB) agg[i] *= 1.0f / fmaxf(deg[i / EMB], 1.0f);
}

// out[a] = relu(dot(out1[p], outW2[s]) + outb2[s]),  p = lab/NSE, s = lab%NSE
__global__ void k_anchor(const float* __restrict__ out1,
                         const float* __restrict__ outW2,
                         const float* __restrict__ outb2,
                         const int* __restrict__ labels,
                         float* __restrict__ out) {
    const int a = (blockIdx.x * blockDim.x + threadIdx.x) >> 5;
    if (a >= NANCH) return;
    const int lane = threadIdx.x & 31;
    const int lab  = labels[a];
    const int p    = lab / NSE;
    const int sIdx = lab % NSE;
    const float* __restrict__ u = out1  + (size_t)p    * EMB;
    const float* __restrict__ w = outW2 + (size_t)sIdx * EMB;
    float acc = 0.f;
#pragma unroll
    for (int q = 0; q < 4; ++q) acc += u[lane + 32 * q] * w[lane + 32 * q];
#pragma unroll
    for (int off = 16; off; off >>= 1) acc += __shfl_xor(acc, off, 32);
    if (lane == 0) out[a] = fmaxf(acc + outb2[sIdx], 0.f);
}

__global__ void k_zero(float* __restrict__ p, int n) {
    const int i = blockIdx.x * blockDim.x + threadIdx.x;
    if (i < n) p[i] = 0.f;
}

// ---------------------------------------------------------------------------
// Launcher
// ---------------------------------------------------------------------------
extern "C" void kernel_launch(void* const* d_in, const int* in_sizes, int n_in,
                              void* d_out, int out_size, void* d_ws, size_t ws_size,
                              hipStream_t stream) {
    const int*   edge_index = (const int*)  d_in[0];   // (2, NE)
    const float* drugFeat   = (const float*)d_in[1];   // (600, 2048)
    const int*   tpl        = (const int*)  d_in[2];   // (8192, 2)
    const int*   anchors    = (const int*)  d_in[3];   // (8192,)
    const float* W1         = (const float*)d_in[4];   // (128, 2048)
    const float* b1         = (const float*)d_in[5];
    const float* W2         = (const float*)d_in[6];   // (128, 128)
    const float* b2         = (const float*)d_in[7];
    const float* protEmb    = (const float*)d_in[8];   // (19081, 128)
    const float* sageWl     = (const float*)d_in[9];   // (2, 128, 128)
    const float* sagebl     = (const float*)d_in[10];  // (2, 128)
    const float* sageWr     = (const float*)d_in[11];  // (2, 128, 128)
    const float* outW1      = (const float*)d_in[12];  // (128, 256)
    const float* outb1      = (const float*)d_in[13];
    const float* outW2      = (const float*)d_in[14];  // (964, 128)
    const float* outb2      = (const float*)d_in[15];
    float* out = (float*)d_out;

    // Workspace layout (bytes, 256-aligned)
    char* ws = (char*)d_ws;
    float* drugF0 = (float*)(ws);                       // 600*128
    float* drugF  = (float*)(ws + 307200);              // 600*128
    float* agg    = (float*)(ws + 614400);              // 600*128
    float* deg    = (float*)(ws + 921600);              // 600 (contiguous after agg)
    float* xd     = (float*)(ws + 924160);              // 600*128
    float* out1   = (float*)(ws + 1231360);             // 8192*128
    (void)ws_size; (void)in_sizes; (void)n_in; (void)out_size;

    const int* esrc = edge_index;
    const int* edst = edge_index + NE_;

    // 1) re-zero atomic accumulators (agg + deg contiguous: 77400 floats)
    k_zero<<<(77400 + 255) / 256, 256, 0, stream>>>(agg, 77400);

    // 2) drugF0 = relu(drugFeatures @ W1^T + b1)
    k_gemm_relu<<<(ND + 15) / 16, 256, 0, stream>>>(drugFeat, ND, FEAT, W1, b1, drugF0);

    // 3) drugF = relu(drugF0 @ W2^T + b2)
    k_gemm_relu<<<(ND + 15) / 16, 256, 0, stream>>>(drugF0, ND, EMB, W2, b2, drugF);

    // 4) edge scatter-add (only dst < ND matters for the final output)
    k_edges<<<2048, 256, 0, stream>>>(esrc, edst, drugF, protEmb, agg, deg);

    // 5) agg *= 1/max(deg,1)
    k_scale<<<(ND * EMB + 255) / 256, 256, 0, stream>>>(agg, deg);

    // 6) xd = relu(agg @ sageWl[1]^T + drugF @ sageWr[1]^T + sagebl[1])
    //    (reference loop never updates xF -> only last layer's weights matter)
    k_sage<<<(ND + 15) / 16, 256, 0, stream>>>(agg, sageWl + EMB * EMB,
                                               drugF, sageWr + EMB * EMB,
                                               sagebl + EMB, xd);

    // 7) out1 = relu(concat(xd[tpl0], xd[tpl1]) @ outW1^T + outb1)
    k_pair<<<NPAIRS / 16, 256, 0, stream>>>(xd, tpl, outW1, outb1, out1);

    // 8) out[a] = relu(dot(out1[p], outW2[s]) + outb2[s]) at anchor positions only
    k_anchor<<<(NANCH * 32 + 255) / 256, 256, 0, stream>>>(out1, outW2, outb2, anchors, out);
}